// Observation_20950850470027
// MI455X (gfx1250) — compile-verified
//
#include <hip/hip_runtime.h>

#define BS      128
#define T       256
#define CH      1024      // 2*VH == 2*SH == visual/sen feature width
#define LCROSS  512
#define NLOC    128
#define NCAT    2688      // 1024 + 1024 + 512 + 128
#define NFEAT   1024
#define TSPLIT  4
#define TCHUNK  (T / TSPLIT)

typedef float v2f __attribute__((ext_vector_type(2)));
typedef float v4f __attribute__((ext_vector_type(4)));
typedef float v8f __attribute__((ext_vector_type(8)));

// -------- workspace layout (float offsets) --------
#define WS_GPART  512                               // after 128*4 meta ints
#define WS_LPART  (WS_GPART + TSPLIT * BS * CH)
#define WS_LV     (WS_LPART + TSPLIT * BS * CH)
#define WS_X      (WS_LV + BS * CH)

// output offsets (floats)
#define OUT_SEN   ((size_t)BS * T * CH)             // 33554432
#define OUT_FEAT  (OUT_SEN + (size_t)BS * CH)       // 33685504

// X column layout: [0,1024)=sen_fea  [1024,2048)=gv_fea  [2048,2560)=vs  [2560,2688)=loc_fea
#define XCOL_GV   1024
#define XCOL_VS   2048
#define XCOL_LOC  2560

// ---------------- meta: v_len, s, e per batch ----------------
__global__ void meta_kernel(const int* __restrict__ vmask,
                            const float* __restrict__ loc,
                            int* __restrict__ meta) {
    int b = threadIdx.x;
    if (b >= BS) return;
    int vl = 0;
#pragma unroll 8
    for (int t = 0; t < T; ++t) vl += vmask[b * T + t];
    float scale = (float)(vl - 1);
    int s = (int)floorf(loc[2 * b + 0] * scale);
    int e = (int)floorf(loc[2 * b + 1] * scale);
    meta[4 * b + 0] = vl;
    meta[4 * b + 1] = s;
    meta[4 * b + 2] = e;
    meta[4 * b + 3] = 0;
}

// ------- fused stream: copy visual_fea -> out (NT ld/st), partial pooled sums -------
__global__ __launch_bounds__(256) void pool_copy_kernel(
        const float* __restrict__ vis, float* __restrict__ out,
        const int* __restrict__ meta,
        float* __restrict__ gpart, float* __restrict__ lpart) {
    const int b = blockIdx.x, sp = blockIdx.y, c4 = threadIdx.x;  // 256 v4f groups
    const int vl = meta[4 * b], s = meta[4 * b + 1], e = meta[4 * b + 2];
    const int t0 = sp * TCHUNK;
    const v4f* __restrict__ src = (const v4f*)(vis + (size_t)(b * T + t0) * CH) + c4;
    v4f* __restrict__ dst = (v4f*)(out + (size_t)(b * T + t0) * CH) + c4;
    v4f g = {}; v4f ls = {};
#pragma unroll 4
    for (int i = 0; i < TCHUNK; ++i) {
        v4f v = __builtin_nontemporal_load(src + (size_t)i * (CH / 4));
        __builtin_nontemporal_store(v, dst + (size_t)i * (CH / 4));
        int t = t0 + i;
        if (t < vl) g += v;
        if (t >= s && t <= e) ls += v;
    }
    ((v4f*)(gpart + (size_t)(sp * BS + b) * CH))[c4] = g;
    ((v4f*)(lpart + (size_t)(sp * BS + b) * CH))[c4] = ls;
}

// ------- reduce partials -> gv_fea, lv_fea; assemble X; copy sen_fea -------
__global__ __launch_bounds__(256) void reduce_kernel(
        const float* __restrict__ gpart, const float* __restrict__ lpart,
        const int* __restrict__ meta, const float* __restrict__ senf,
        float* __restrict__ lv, float* __restrict__ X, float* __restrict__ out_sen) {
    const int b = blockIdx.x, c4 = threadIdx.x;
    const int vl = meta[4 * b], s = meta[4 * b + 1], e = meta[4 * b + 2];
    v4f g = {}; v4f ls = {};
#pragma unroll
    for (int sp = 0; sp < TSPLIT; ++sp) {
        g  += ((const v4f*)(gpart + (size_t)(sp * BS + b) * CH))[c4];
        ls += ((const v4f*)(lpart + (size_t)(sp * BS + b) * CH))[c4];
    }
    const float invv = 1.f / (float)vl;
    v4f gv = g * invv;
    v4f lvv = {};
    if (s <= e) {
        int cnt = e + 1 - s; if (cnt < 1) cnt = 1;
        lvv = ls * (1.f / (float)cnt);
    }
    ((v4f*)(lv + (size_t)b * CH))[c4] = lvv;
    ((v4f*)(X + (size_t)b * NCAT + XCOL_GV))[c4] = gv;               // gv_fea cols [1024,2048)
    v4f sf = ((const v4f*)(senf + (size_t)b * CH))[c4];
    ((v4f*)(X + (size_t)b * NCAT))[c4] = sf;                         // sen_fea cols [0,1024)
    __builtin_nontemporal_store(sf, (v4f*)(out_sen + (size_t)b * CH) + c4);  // output copy
}

// ------------- loc head: relu(loc @ W3^T + b3), K = 2 -------------
__global__ void loc_kernel(const float* __restrict__ loc, const float* __restrict__ W3,
                           const float* __restrict__ b3, float* __restrict__ X) {
    const int b = blockIdx.x, n = threadIdx.x;
    float acc = fmaf(loc[2 * b], W3[2 * n], fmaf(loc[2 * b + 1], W3[2 * n + 1], b3[n]));
    X[(size_t)b * NCAT + XCOL_LOC + n] = acc > 0.f ? acc : 0.f;
}

// ------- fused dual WMMA GEMM: vs = relu((lv@W1^T+b1)*(sen@W2^T+b2)) -------
// M=128, N=512, K=1024; one 16x16 tile per wave, V_WMMA_F32_16X16X4_F32.
__global__ __launch_bounds__(256) void vs_gemm_kernel(
        const float* __restrict__ lv, const float* __restrict__ senf,
        const float* __restrict__ W1, const float* __restrict__ b1,
        const float* __restrict__ W2, const float* __restrict__ b2,
        float* __restrict__ X) {
    const int wave = (blockIdx.x * 256 + threadIdx.x) >> 5;   // 256 waves total
    const int lane = threadIdx.x & 31;
    const int l = lane & 15, hi = lane >> 4;
    const int tileN = wave & 31;           // 512/16
    const int tileM = wave >> 5;           // 8
    const int row0 = tileM * 16, col0 = tileN * 16;
    const float* __restrict__ a1 = lv   + (size_t)(row0 + l) * CH;
    const float* __restrict__ a2 = senf + (size_t)(row0 + l) * CH;
    const float* __restrict__ w1 = W1   + (size_t)(col0 + l) * CH;
    const float* __restrict__ w2 = W2   + (size_t)(col0 + l) * CH;
    v8f c1 = {}; v8f c2 = {};
#pragma unroll 4
    for (int k0 = 0; k0 < CH; k0 += 4) {
        const int ka = k0 + 2 * hi;
        v2f fa1 = *(const v2f*)(a1 + ka);
        v2f fb1 = *(const v2f*)(w1 + ka);
        v2f fa2 = *(const v2f*)(a2 + ka);
        v2f fb2 = *(const v2f*)(w2 + ka);
        c1 = __builtin_amdgcn_wmma_f32_16x16x4_f32(false, fa1, false, fb1, (short)0, c1, false, false);
        c2 = __builtin_amdgcn_wmma_f32_16x16x4_f32(false, fa2, false, fb2, (short)0, c2, false, false);
    }
    const float bb1 = b1[col0 + l], bb2 = b2[col0 + l];
#pragma unroll
    for (int r = 0; r < 8; ++r) {
        float y = (c1[r] + bb1) * (c2[r] + bb2);
        y = y > 0.f ? y : 0.f;
        X[(size_t)(row0 + r + 8 * hi) * NCAT + XCOL_VS + col0 + l] = y;   // vs cols [2048,2560)
    }
}

// ------- final WMMA GEMM: feature = relu(X @ W4^T + b4), M=128 N=1024 K=2688 -------
__global__ __launch_bounds__(256) void feat_gemm_kernel(
        const float* __restrict__ X, const float* __restrict__ W4,
        const float* __restrict__ b4, float* __restrict__ outf) {
    const int wave = (blockIdx.x * 256 + threadIdx.x) >> 5;   // 512 waves total
    const int lane = threadIdx.x & 31;
    const int l = lane & 15, hi = lane >> 4;
    const int tileN = wave & 63;           // 1024/16
    const int tileM = wave >> 6;           // 8
    const int row0 = tileM * 16, col0 = tileN * 16;
    const float* __restrict__ a = X  + (size_t)(row0 + l) * NCAT;
    const float* __restrict__ w = W4 + (size_t)(col0 + l) * NCAT;
    v8f c = {};
#pragma unroll 4
    for (int k0 = 0; k0 < NCAT; k0 += 4) {
        const int ka = k0 + 2 * hi;
        v2f fa = *(const v2f*)(a + ka);
        v2f fb = *(const v2f*)(w + ka);
        c = __builtin_amdgcn_wmma_f32_16x16x4_f32(false, fa, false, fb, (short)0, c, false, false);
    }
    const float bb = b4[col0 + l];
#pragma unroll
    for (int r = 0; r < 8; ++r) {
        float y = c[r] + bb;
        y = y > 0.f ? y : 0.f;
        outf[(size_t)(row0 + r + 8 * hi) * NFEAT + col0 + l] = y;
    }
}

extern "C" void kernel_launch(void* const* d_in, const int* in_sizes, int n_in,
                              void* d_out, int out_size, void* d_ws, size_t ws_size,
                              hipStream_t stream) {
    const float* vis   = (const float*)d_in[3];
    const float* senf  = (const float*)d_in[4];
    const float* loc   = (const float*)d_in[5];
    const int*   vmask = (const int*)d_in[7];
    const float* W1 = (const float*)d_in[8];
    const float* b1 = (const float*)d_in[9];
    const float* W2 = (const float*)d_in[10];
    const float* b2 = (const float*)d_in[11];
    const float* W3 = (const float*)d_in[12];
    const float* b3 = (const float*)d_in[13];
    const float* W4 = (const float*)d_in[14];
    const float* b4 = (const float*)d_in[15];

    float* out = (float*)d_out;
    float* wsf = (float*)d_ws;
    int*   meta  = (int*)d_ws;
    float* gpart = wsf + WS_GPART;
    float* lpart = wsf + WS_LPART;
    float* lv    = wsf + WS_LV;
    float* X     = wsf + WS_X;

    meta_kernel<<<1, 128, 0, stream>>>(vmask, loc, meta);
    pool_copy_kernel<<<dim3(BS, TSPLIT), 256, 0, stream>>>(vis, out, meta, gpart, lpart);
    reduce_kernel<<<BS, 256, 0, stream>>>(gpart, lpart, meta, senf, lv, X, out + OUT_SEN);
    loc_kernel<<<BS, NLOC, 0, stream>>>(loc, W3, b3, X);
    vs_gemm_kernel<<<32, 256, 0, stream>>>(lv, senf, W1, b1, W2, b2, X);
    feat_gemm_kernel<<<64, 256, 0, stream>>>(X, W4, b4, out + OUT_FEAT);
}